// GCN_52656299049561
// MI455X (gfx1250) — compile-verified
//
#include <hip/hip_runtime.h>
#include <hip/hip_bf16.h>

typedef __attribute__((ext_vector_type(2))) float v2f;
typedef __attribute__((ext_vector_type(8))) float v8f;

#define N0_   20000
#define N1_   15000
#define N2_   15000
#define NTOT  50000
#define NE    800000
#define HDIM  128
#define CDIM  64

// ---------------- elementwise helpers ----------------

__global__ void zero_f32(float* __restrict__ p, int n) {
    int i = blockIdx.x * blockDim.x + threadIdx.x;
    if (i < n) p[i] = 0.0f;
}

__global__ void degree_kernel(const int* __restrict__ src, const int* __restrict__ dst,
                              float* __restrict__ deg_out, float* __restrict__ deg_in, int e) {
    int i = blockIdx.x * blockDim.x + threadIdx.x;
    if (i < e) {
        atomicAdd(&deg_out[src[i]], 1.0f);
        atomicAdd(&deg_in[dst[i]], 1.0f);
    }
}

// in-place: deg -> max(deg,1)^-0.5  (exact sqrt+div for accuracy)
__global__ void norm_kernel(float* __restrict__ p, int n) {
    int i = blockIdx.x * blockDim.x + threadIdx.x;
    if (i < n) p[i] = 1.0f / sqrtf(fmaxf(p[i], 1.0f));
}

// a[dst] += h[src] * ns[src]; 32 threads per edge, float4 per thread (H=128)
__global__ void edge_agg(const float* __restrict__ h, const int* __restrict__ src,
                         const int* __restrict__ dst, const float* __restrict__ ns,
                         float* __restrict__ acc, int e) {
    int idx = blockIdx.x * blockDim.x + threadIdx.x;
    int ed = idx >> 5;
    if (ed >= e) return;
    int c = (idx & 31) << 2;
    int s = src[ed];
    int d = dst[ed];
    float sc = ns[s];
    const float4 v = *reinterpret_cast<const float4*>(h + (size_t)s * HDIM + c);
    float* o = acc + (size_t)d * HDIM + c;
    atomicAdd(o + 0, v.x * sc);
    atomicAdd(o + 1, v.y * sc);
    atomicAdd(o + 2, v.z * sc);
    atomicAdd(o + 3, v.w * sc);
}

// h = relu(a * nd[row] + bias[col])   (layer 0 epilogue)
__global__ void layer0_post(const float* __restrict__ a, const float* __restrict__ nd,
                            const float* __restrict__ bias, float* __restrict__ h, int n) {
    int i = blockIdx.x * blockDim.x + threadIdx.x;
    if (i < n) {
        int row = i >> 7;              // H = 128
        int col = i & (HDIM - 1);
        float v = a[i] * nd[row] + bias[col];
        h[i] = fmaxf(v, 0.0f);
    }
}

// ---------------- fp32 WMMA GEMM ----------------
// Y[M x HOUT] = (optional relu)( (rowscale ? diag(rowscale):I) * X[M x K] @ W[K x HOUT] + bias )
// One wave32 computes a 16-row strip across all HOUT columns.
// A layout (V_WMMA_F32_16X16X4_F32, 16x4 fp32): lanes 0-15 hold M=lane, VGPR0/1 = K0/K1;
// lanes 16-31 hold M=lane-16, VGPR0/1 = K2/K3. B (4x16) symmetric with N in lanes.
// C/D: VGPR r -> M = r (lanes 0-15) / r+8 (lanes 16-31), N = lane&15.
template <int HOUT, bool RELU>
__global__ void gemm_wmma_f32(const float* __restrict__ X, const float* __restrict__ W,
                              const float* __restrict__ bias, const float* __restrict__ rowscale,
                              float* __restrict__ Y, int M, int K) {
    constexpr int NT = HOUT / 16;
    int strip = blockIdx.x * blockDim.y + threadIdx.y;   // uniform per wave
    int row0 = strip * 16;
    if (row0 >= M) return;                               // wave-uniform exit

    int lane = threadIdx.x;        // 0..31
    int lo   = lane & 15;
    int hi   = lane >> 4;          // 0 or 1

    int  rowA = row0 + lo;
    bool okA  = rowA < M;
    // zero-scale out-of-range rows instead of predicating -> EXEC stays all-1s for WMMA
    float rs = okA ? (rowscale ? rowscale[rowA] : 1.0f) : 0.0f;
    const float* xrow = X + (size_t)(okA ? rowA : 0) * K;

    v8f acc[NT] = {};

    for (int k = 0; k < K; k += 4) {
        int kb = k + hi * 2;
        v2f av;
        av.x = xrow[kb]     * rs;
        av.y = xrow[kb + 1] * rs;
#pragma unroll
        for (int t = 0; t < NT; ++t) {
            v2f bv;
            bv.x = W[(size_t)kb       * HOUT + t * 16 + lo];
            bv.y = W[(size_t)(kb + 1) * HOUT + t * 16 + lo];
            acc[t] = __builtin_amdgcn_wmma_f32_16x16x4_f32(
                /*neg_a=*/false, av, /*neg_b=*/false, bv,
                /*c_mod=*/(short)0, acc[t], /*reuse_a=*/false, /*reuse_b=*/false);
        }
    }

#pragma unroll
    for (int t = 0; t < NT; ++t) {
        int col = t * 16 + lo;
        float bb = bias[col];
#pragma unroll
        for (int r = 0; r < 8; ++r) {
            int row = row0 + r + hi * 8;
            if (row < M) {
                float v = acc[t][r] + bb;
                if (RELU) v = fmaxf(v, 0.0f);
                Y[(size_t)row * HOUT + col] = v;
            }
        }
    }
}

// ---------------- host driver ----------------

extern "C" void kernel_launch(void* const* d_in, const int* in_sizes, int n_in,
                              void* d_out, int out_size, void* d_ws, size_t ws_size,
                              hipStream_t stream) {
    (void)in_sizes; (void)n_in; (void)out_size; (void)ws_size;

    const float* feat0   = (const float*)d_in[0];
    const float* feat1   = (const float*)d_in[1];
    const float* feat2   = (const float*)d_in[2];
    const float* W0      = (const float*)d_in[3];
    const float* b0      = (const float*)d_in[4];
    const float* W1      = (const float*)d_in[5];
    const float* b1      = (const float*)d_in[6];
    const float* W2      = (const float*)d_in[7];
    const float* b2      = (const float*)d_in[8];
    const float* bias_l0 = (const float*)d_in[9];
    const float* W_l1    = (const float*)d_in[10];
    const float* b_l1    = (const float*)d_in[11];
    const float* W_l2    = (const float*)d_in[12];
    const float* b_l2    = (const float*)d_in[13];
    const int*   src     = (const int*)d_in[14];
    const int*   dst     = (const int*)d_in[15];
    float*       out     = (float*)d_out;

    // workspace layout (floats): h[N*H] | a[N*H] | ns[N] | nd[N]
    float* ws = (float*)d_ws;
    float* h  = ws;
    float* a  = ws + (size_t)NTOT * HDIM;
    float* ns = ws + 2 * (size_t)NTOT * HDIM;
    float* nd = ns + NTOT;

    const int BZ = 256;
    const int nh = NTOT * HDIM;                 // 6,400,000
    dim3 gwave(32, 4);                          // 4 waves / block, 1 strip each

    // ---- degrees -> norms ----
    zero_f32<<<(2 * NTOT + BZ - 1) / BZ, BZ, 0, stream>>>(ns, 2 * NTOT);
    degree_kernel<<<(NE + BZ - 1) / BZ, BZ, 0, stream>>>(src, dst, ns, nd, NE);
    norm_kernel<<<(2 * NTOT + BZ - 1) / BZ, BZ, 0, stream>>>(ns, 2 * NTOT);

    // ---- per-type projections -> h ----
    {
        int s0 = (N0_ + 15) / 16;
        gemm_wmma_f32<HDIM, false><<<(s0 + 3) / 4, gwave, 0, stream>>>(
            feat0, W0, b0, nullptr, h, N0_, 512);
        int s1 = (N1_ + 15) / 16;
        gemm_wmma_f32<HDIM, false><<<(s1 + 3) / 4, gwave, 0, stream>>>(
            feat1, W1, b1, nullptr, h + (size_t)N0_ * HDIM, N1_, 256);
        int s2 = (N2_ + 15) / 16;
        gemm_wmma_f32<HDIM, false><<<(s2 + 3) / 4, gwave, 0, stream>>>(
            feat2, W2, b2, nullptr, h + (size_t)(N0_ + N1_) * HDIM, N2_, 128);
    }

    const int eagg_grid = (NE * 32 + BZ - 1) / BZ;

    // ---- layer 0: agg + bias + relu ----
    zero_f32<<<(nh + BZ - 1) / BZ, BZ, 0, stream>>>(a, nh);
    edge_agg<<<eagg_grid, BZ, 0, stream>>>(h, src, dst, ns, a, NE);
    layer0_post<<<(nh + BZ - 1) / BZ, BZ, 0, stream>>>(a, nd, bias_l0, h, nh);

    // ---- layer 1: agg, (a*nd) @ W_l1 + b, relu ----
    zero_f32<<<(nh + BZ - 1) / BZ, BZ, 0, stream>>>(a, nh);
    edge_agg<<<eagg_grid, BZ, 0, stream>>>(h, src, dst, ns, a, NE);
    {
        int s = (NTOT + 15) / 16;
        gemm_wmma_f32<HDIM, true><<<(s + 3) / 4, gwave, 0, stream>>>(
            a, W_l1, b_l1, nd, h, NTOT, HDIM);
    }

    // ---- layer 2: agg, (a*nd) @ W_l2 + b -> out ----
    zero_f32<<<(nh + BZ - 1) / BZ, BZ, 0, stream>>>(a, nh);
    edge_agg<<<eagg_grid, BZ, 0, stream>>>(h, src, dst, ns, a, NE);
    {
        int s = (NTOT + 15) / 16;
        gemm_wmma_f32<CDIM, false><<<(s + 3) / 4, gwave, 0, stream>>>(
            a, W_l2, b_l2, nd, out, NTOT, HDIM);
    }
}